// GraphScaledDotProductAttentionWithMask_37134287242044
// MI455X (gfx1250) — compile-verified
//
#include <hip/hip_runtime.h>
#include <hip/hip_bf16.h>
#include <math.h>

// Problem constants (from reference)
#define B_  4
#define H_  16
#define LQ_ 1024
#define LK_ 1024
#define DK_ 64
#define DV_ 64
#define DM_ 1024
#define QT  32          // query rows per attention block (LDS tile)

typedef __attribute__((ext_vector_type(2))) float v2f;
typedef __attribute__((ext_vector_type(8))) float v8f;

static __device__ __forceinline__ v8f wmma4(v2f a, v2f b, v8f c) {
  // V_WMMA_F32_16X16X4_F32: D = A(16x4) * B(4x16) + C(16x16), fp32
  // 8-arg pattern: (neg_a, A, neg_b, B, c_mod, C, reuse_a, reuse_b)
  return __builtin_amdgcn_wmma_f32_16x16x4_f32(
      false, a, false, b, (short)0, c, false, false);
}

// ---------------------------------------------------------------------------
// Kernel 0: per-query positional MLP -> pos in [0, LK-1]
//   pos = sigmoid( tanh(scaled_q @ Wv + bv) @ Ws + bs ) * (LK-1)
// One thread per (b,h,q). Weights staged in LDS.
// ---------------------------------------------------------------------------
__global__ void pos_kernel(const float* __restrict__ q,
                           const float* __restrict__ Wv,
                           const float* __restrict__ bv,
                           const float* __restrict__ Ws,
                           const float* __restrict__ bs,
                           float* __restrict__ pos_out) {
  __shared__ float sW[DK_ * DV_];
  __shared__ float sWs[DV_];
  __shared__ float sbv[DV_];
  __shared__ float sbs;
  for (int i = threadIdx.x; i < DK_ * DV_; i += blockDim.x) sW[i] = Wv[i];
  for (int i = threadIdx.x; i < DV_; i += blockDim.x) { sWs[i] = Ws[i]; sbv[i] = bv[i]; }
  if (threadIdx.x == 0) sbs = bs[0];
  __syncthreads();

  const int idx = blockIdx.x * blockDim.x + threadIdx.x;   // 0 .. B*H*LQ-1
  const float scale = 0.125f;                              // 1/sqrt(64)
  const float* qr = q + (size_t)idx * DK_;
  float sq[DK_];
#pragma unroll
  for (int d = 0; d < DK_; ++d) sq[d] = qr[d] * scale;

  float s = sbs;
  for (int j = 0; j < DV_; ++j) {
    float acc = sbv[j];
#pragma unroll
    for (int d = 0; d < DK_; ++d) acc = fmaf(sq[d], sW[d * DV_ + j], acc);
    s = fmaf(tanhf(acc), sWs[j], s);
  }
  const float sig = 1.0f / (1.0f + __expf(-s));
  pos_out[idx] = sig * (float)(LK_ - 1);
}

// ---------------------------------------------------------------------------
// Kernel 1: fused attention for one (b,h, 32-query tile).
//   S = scaledQ @ K^T + bias + gaussian(graph_attn_bias gather/lerp)
//   P = softmax(S)  (kept in LDS, also written to `weights` output)
//   O = P @ V       (written to ws as [B, LQ, H, DV] == [B, LQ, DM])
// 256 threads = 8 wave32, logits tile 32x1024 fp32 resident in LDS (~137KB).
// ---------------------------------------------------------------------------
__global__ void attn_kernel(const float* __restrict__ q,
                            const float* __restrict__ k,
                            const float* __restrict__ v,
                            const float* __restrict__ bias,
                            const float* __restrict__ gab,
                            const float* __restrict__ pos,
                            float* __restrict__ weights_out,
                            float* __restrict__ o_out) {
  constexpr int SQ_STRIDE = DK_ + 2;   // 66  (bank-conflict padding)
  constexpr int SP_STRIDE = LK_ + 2;   // 1026
  extern __shared__ float smem[];
  float* sQ  = smem;                          // QT * SQ_STRIDE
  float* sP  = sQ + QT * SQ_STRIDE;           // QT * SP_STRIDE
  float* sWU = sP + QT * SP_STRIDE;           // QT
  float* sWD = sWU + QT;                      // QT
  int*   sIU = (int*)(sWD + QT);              // QT
  int*   sID = sIU + QT;                      // QT

  const int tid  = threadIdx.x;
  const int lane = tid & 31;
  const int wave = tid >> 5;
  const int l16  = lane & 15;
  const int hf   = lane >> 4;

  const int bh    = blockIdx.x / (LQ_ / QT);
  const int qt    = blockIdx.x % (LQ_ / QT);
  const int qbase = qt * QT;
  const float scale = 0.125f;

  // ---- stage scaled Q tile ----
  for (int jj = tid; jj < QT * DK_; jj += blockDim.x) {
    const int row = jj >> 6, d = jj & 63;
    sQ[row * SQ_STRIDE + d] =
        q[((size_t)bh * LQ_ + qbase + row) * DK_ + d] * scale;
  }
  // ---- per-row gaussian interpolation factors ----
  if (tid < QT) {
    const float p  = pos[(size_t)bh * LQ_ + qbase + tid];
    const float cu = ceilf(p), fd = floorf(p);
    sIU[tid] = (int)cu;
    sID[tid] = (int)fd;
    sWU[tid] = 1.0f - (cu - p);
    sWD[tid] = 1.0f - (p - fd);
  }
  __syncthreads();

  // ---- S = Q K^T (WMMA f32 16x16x4), + bias + gaussian, into LDS ----
  const float* kbh    = k    + (size_t)bh * LK_ * DK_;
  const float* gabbh  = gab  + (size_t)bh * LK_ * LK_;
  const float* biasbh = bias + (size_t)bh * LQ_ * LK_;

  for (int nt = wave; nt < LK_ / 16; nt += 8) {
    const int ncol = nt * 16 + l16;                 // this lane's key column
    const float* krow = kbh + (size_t)ncol * DK_;   // K row == B-matrix column
    if (nt + 8 < LK_ / 16)
      __builtin_prefetch(kbh + (size_t)((nt + 8) * 16 + l16) * DK_, 0, 1);

    v2f bfrag[16];
#pragma unroll
    for (int ks = 0; ks < 16; ++ks)
      bfrag[ks] = *(const v2f*)(krow + 4 * ks + 2 * hf);

#pragma unroll
    for (int mt = 0; mt < QT / 16; ++mt) {
      v8f c = {};
#pragma unroll
      for (int ks = 0; ks < 16; ++ks) {
        v2f a = *(const v2f*)(sQ + (mt * 16 + l16) * SQ_STRIDE + 4 * ks + 2 * hf);
        c = wmma4(a, bfrag[ks], c);
      }
#pragma unroll
      for (int r = 0; r < 8; ++r) {
        const int m  = mt * 16 + r + 8 * hf;        // local query row
        const int qg = qbase + m;
        float val = c[r] + biasbh[(size_t)qg * LK_ + ncol];
        const float sel =
            sWU[m] * gabbh[(size_t)sIU[m] * LK_ + ncol] +
            sWD[m] * gabbh[(size_t)sID[m] * LK_ + ncol];
        sP[m * SP_STRIDE + ncol] = val - 0.5f * sel * sel;
      }
    }
  }
  __syncthreads();

  // ---- softmax rows in LDS; emit normalized weights to HBM ----
  float* wout = weights_out + (size_t)bh * LQ_ * LK_;
  for (int row = wave * 4; row < wave * 4 + 4; ++row) {
    float mx = -3.402823466e38f;
    for (int j = lane; j < LK_; j += 32)
      mx = fmaxf(mx, sP[row * SP_STRIDE + j]);
#pragma unroll
    for (int off = 16; off > 0; off >>= 1)
      mx = fmaxf(mx, __shfl_xor(mx, off, 32));

    float sum = 0.0f;
    for (int j = lane; j < LK_; j += 32) {
      const float e = __expf(sP[row * SP_STRIDE + j] - mx);
      sP[row * SP_STRIDE + j] = e;
      sum += e;
    }
#pragma unroll
    for (int off = 16; off > 0; off >>= 1)
      sum += __shfl_xor(sum, off, 32);

    const float inv = 1.0f / sum;
    for (int j = lane; j < LK_; j += 32) {
      const float p = sP[row * SP_STRIDE + j] * inv;
      sP[row * SP_STRIDE + j] = p;
      wout[(size_t)(qbase + row) * LK_ + j] = p;
    }
  }
  __syncthreads();

  // ---- O = P @ V : one 16x16 tile per wave (2 mt x 4 nt = 8 tiles) ----
  {
    const int mt = wave >> 2;
    const int nt = wave & 3;
    const float* vbh = v + (size_t)bh * LK_ * DV_;
    const int d = nt * 16 + l16;

    v8f c = {};
    for (int ks = 0; ks < LK_ / 4; ++ks) {
      const int kk = 4 * ks + 2 * hf;
      if ((ks & 15) == 0 && ks + 16 < LK_ / 4)
        __builtin_prefetch(vbh + (size_t)(4 * (ks + 16)) * DV_ + d, 0, 1);
      v2f a = *(const v2f*)(sP + (mt * 16 + l16) * SP_STRIDE + kk);
      v2f b;
      b.x = vbh[(size_t)kk * DV_ + d];
      b.y = vbh[(size_t)(kk + 1) * DV_ + d];
      c = wmma4(a, b, c);
    }
    const int bi = bh / H_, hi = bh % H_;
#pragma unroll
    for (int r = 0; r < 8; ++r) {
      const int qg = qbase + mt * 16 + r + 8 * hf;
      o_out[(((size_t)bi * LQ_ + qg) * H_ + hi) * DV_ + d] = c[r];
    }
  }
}

// ---------------------------------------------------------------------------
// Kernel 2: graph_out = O[B*LQ, DM] @ W_out[DM, DM] + b_out  (WMMA f32)
// Block = 64x64 output tile, 8 waves, 2 accum tiles per wave.
// ---------------------------------------------------------------------------
__global__ void out_gemm_kernel(const float* __restrict__ X,
                                const float* __restrict__ W,
                                const float* __restrict__ bo,
                                float* __restrict__ out) {
  const int tid  = threadIdx.x;
  const int lane = tid & 31;
  const int wave = tid >> 5;
  const int l16  = lane & 15;
  const int hf   = lane >> 4;

  const int M2 = B_ * LQ_;                 // 4096
  const int bm = blockIdx.x % (M2 / 64);
  const int bn = blockIdx.x / (M2 / 64);

  const int mt  = wave & 3;
  const int ntg = (wave >> 2) * 2;
  const int mrow = bm * 64 + mt * 16 + l16;
  const int n0   = bn * 64 + ntg * 16 + l16;
  const int n1   = n0 + 16;

  const float* xrow = X + (size_t)mrow * DM_;
  v8f c0 = {}, c1 = {};
  for (int ks = 0; ks < DM_ / 4; ++ks) {
    const int kk = 4 * ks + 2 * hf;
    if ((ks & 15) == 0 && ks + 16 < DM_ / 4)
      __builtin_prefetch(xrow + 4 * (ks + 16), 0, 1);
    v2f a = *(const v2f*)(xrow + kk);
    const float* w0 = W + (size_t)kk * DM_;
    v2f b0, b1;
    b0.x = w0[n0];        b0.y = w0[DM_ + n0];
    b1.x = w0[n1];        b1.y = w0[DM_ + n1];
    c0 = wmma4(a, b0, c0);
    c1 = wmma4(a, b1, c1);
  }
  const float bb0 = bo[n0], bb1 = bo[n1];
#pragma unroll
  for (int r = 0; r < 8; ++r) {
    const int m = bm * 64 + mt * 16 + r + 8 * hf;
    out[(size_t)m * DM_ + n0] = c0[r] + bb0;
    out[(size_t)m * DM_ + n1] = c1[r] + bb1;
  }
}

// ---------------------------------------------------------------------------
extern "C" void kernel_launch(void* const* d_in, const int* in_sizes, int n_in,
                              void* d_out, int out_size, void* d_ws, size_t ws_size,
                              hipStream_t stream) {
  (void)in_sizes; (void)n_in; (void)out_size; (void)ws_size;

  const float* q    = (const float*)d_in[0];
  const float* k    = (const float*)d_in[1];
  const float* v    = (const float*)d_in[2];
  const float* bias = (const float*)d_in[3];
  const float* gab  = (const float*)d_in[4];
  const float* Wv   = (const float*)d_in[5];
  const float* bv   = (const float*)d_in[6];
  const float* Ws   = (const float*)d_in[7];
  const float* bs   = (const float*)d_in[8];
  const float* Wo   = (const float*)d_in[9];
  const float* bo   = (const float*)d_in[10];

  float* out       = (float*)d_out;
  float* graph_out = out;                              // [B, LQ, DM]
  float* weights   = out + (size_t)B_ * LQ_ * DM_;     // [B, H, LQ, LK]

  float* ws_pos = (float*)d_ws;                        // B*H*LQ floats
  float* ws_O   = ws_pos + (size_t)B_ * H_ * LQ_;      // B*LQ*DM floats (16MB)

  // K0: positional MLP
  pos_kernel<<<(B_ * H_ * LQ_) / 256, 256, 0, stream>>>(q, Wv, bv, Ws, bs, ws_pos);

  // K1: fused attention (dynamic LDS ~137KB of the 320KB/WGP)
  const size_t smem_bytes =
      (size_t)(QT * (DK_ + 2) + QT * (LK_ + 2) + 2 * QT) * sizeof(float) +
      (size_t)(2 * QT) * sizeof(int);
  (void)hipFuncSetAttribute((const void*)attn_kernel,
                            hipFuncAttributeMaxDynamicSharedMemorySize,
                            (int)smem_bytes);
  attn_kernel<<<B_ * H_ * (LQ_ / QT), 256, smem_bytes, stream>>>(
      q, k, v, bias, gab, ws_pos, weights, ws_O);

  // K2: output projection
  out_gemm_kernel<<<(B_ * LQ_ / 64) * (DM_ / 64), 256, 0, stream>>>(
      ws_O, Wo, bo, graph_out);
}